// Hist_net_44916767981882
// MI455X (gfx1250) — compile-verified
//
#include <hip/hip_runtime.h>
#include <hip/hip_bf16.h>

typedef __attribute__((ext_vector_type(2))) float v2f;
typedef __attribute__((ext_vector_type(8))) float v8f;

// ---------------------------------------------------------------------------
// Kernel 1: histogram.  img [64,3,512,512] f32 (integer values 0..255).
// Per (b,c) slice: 512*512 = 262144 pixels -> 256 bins.
// Grid = 192 slices * 8 splits = 1536 blocks, 256 threads.
// Per-wave private LDS histograms (8 waves * 256 bins u32) -> ds_add_u32,
// then one global_atomic_add_u32 per bin per block (deterministic: integers).
// ---------------------------------------------------------------------------
#define HSPLIT 8

__global__ __launch_bounds__(256) void hist_kernel(
    const float* __restrict__ img, unsigned* __restrict__ ghist) {
  __shared__ unsigned lh[8 * 256];
  const int tid = threadIdx.x;
#pragma unroll
  for (int i = tid; i < 8 * 256; i += 256) lh[i] = 0u;
  __syncthreads();

  const int slice = blockIdx.x / HSPLIT;   // 0..191  == b*3 + c
  const int part  = blockIdx.x % HSPLIT;
  const float4* p =
      (const float4*)(img + (size_t)slice * 262144) + (size_t)part * 8192;
  unsigned* mh = lh + ((tid >> 5) << 8);   // per-wave histogram copy

  const float scale = 256.0f / 255.0f;
#pragma unroll 4
  for (int it = 0; it < 32; ++it) {
    float4 v = p[it * 256 + tid];
    int b0 = (int)floorf(v.x * scale); b0 = b0 < 0 ? 0 : (b0 > 255 ? 255 : b0);
    int b1 = (int)floorf(v.y * scale); b1 = b1 < 0 ? 0 : (b1 > 255 ? 255 : b1);
    int b2 = (int)floorf(v.z * scale); b2 = b2 < 0 ? 0 : (b2 > 255 ? 255 : b2);
    int b3 = (int)floorf(v.w * scale); b3 = b3 < 0 ? 0 : (b3 > 255 ? 255 : b3);
    atomicAdd(&mh[b0], 1u);
    atomicAdd(&mh[b1], 1u);
    atomicAdd(&mh[b2], 1u);
    atomicAdd(&mh[b3], 1u);
  }
  __syncthreads();

  unsigned t = 0;
#pragma unroll
  for (int w = 0; w < 8; ++w) t += lh[w * 256 + tid];
  if (t) atomicAdd(&ghist[slice * 256 + tid], t);
}

// ---------------------------------------------------------------------------
// Kernel 2: whole 4-stage SHFC net in ONE 1024-thread workgroup.
// ALL intermediate tensors are channel-padded to 16: [64][16][256] f32, with
// pad channels kept at exactly 0.  This makes every WMMA store unconditional
// (pad rows of A are 0 => acc 0, pad bias preloaded 0 => stores 0) and lets
// the im2col loads skip the channel-range predicate entirely.
// Convs: D(16x16) = A(16 x KPAD zero-padded weights) x B(im2col tile) with
// v_wmma_f32_16x16x4_f32.  BN uses training-mode batch stats (N*L = 16384).
// ---------------------------------------------------------------------------
struct StageParams {
  const float *se_w, *se_b, *se_g, *se_bb;
  const float *hdf_w, *hdf_b, *hdf_g, *hdf_bb;
  const float *comp_w, *comp_b, *comp_g, *comp_bb;
  const float *pw_w, *pw_b, *pw_g, *pw_bb;
};
struct NetParams { StageParams s[4]; };

__device__ __forceinline__ void gsync() {
  __threadfence();   // make global writes visible WGP-wide (L0 inv / L2)
  __syncthreads();
}

// B element: B[(ic,tap)][col] = Xn[ic][col+tap-pad].  Channel index always
// lands inside the zeroed 16-channel padding, so only the column clamp is
// needed (folds to a plain load for TAPS==1).
template <int TAPS>
__device__ __forceinline__ float loadB(const float* __restrict__ Xn, int kk,
                                       int col) {
  const int ic  = kk / TAPS;
  const int t   = kk % TAPS;
  const int src = col + t - (TAPS == 3 ? 1 : 0);
  if (TAPS == 1) return Xn[ic * 256 + src];
  const bool ok  = (unsigned)src < 256u;
  const int srcc = src < 0 ? 0 : (src > 255 ? 255 : src);
  const float v  = Xn[ic * 256 + srcc];
  return ok ? v : 0.0f;
}

// One grouped conv as GEMM on padded tensors [64][16][256].
// Wave-level tasks: (sample n, 16-col tile) -> 64*16 = 1024 tasks / 32 waves.
template <int TAPS, int KPAD, int MREAL>
__device__ void conv_wmma(const float* __restrict__ Xin,
                          float* __restrict__ Yout,
                          const float* __restrict__ bias,
                          const float* __restrict__ Alds) {
  constexpr int NKS = KPAD / 4;
  const int tid  = threadIdx.x;
  const int wave = tid >> 5;
  const int lane = tid & 31;
  const int half = lane >> 4;    // ISA A/B frag: lanes16-31 hold K+2/K+3
  const int mn   = lane & 15;    // A: row M; B: column N
  const int kh   = half << 1;

  // Task-invariant A-fragments and bias: hoist out of the task loop.
  v2f afrag[NKS];
#pragma unroll
  for (int ks = 0; ks < NKS; ++ks) {
    afrag[ks].x = Alds[mn * KPAD + ks * 4 + kh];
    afrag[ks].y = Alds[mn * KPAD + ks * 4 + kh + 1];
  }
  float bv[8];
#pragma unroll
  for (int r = 0; r < 8; ++r) {
    const int m = r + (half << 3);
    bv[r] = (m < MREAL) ? bias[m] : 0.0f;   // pad rows get bias 0
  }

  for (int task = wave; task < 64 * 16; task += 32) {
    const int n  = task >> 4;
    const int l0 = (task & 15) << 4;
    const float* Xn = Xin + (n << 12);      // n * 16ch * 256
    const int col = l0 + mn;

    // Issue all B loads up front so they are in flight together.
    v2f bfrag[NKS];
#pragma unroll
    for (int ks = 0; ks < NKS; ++ks) {
      const int kk0 = ks * 4 + kh;
      bfrag[ks].x = loadB<TAPS>(Xn, kk0, col);
      bfrag[ks].y = loadB<TAPS>(Xn, kk0 + 1, col);
    }

    v8f acc = {};
#pragma unroll
    for (int ks = 0; ks < NKS; ++ks) {
      acc = __builtin_amdgcn_wmma_f32_16x16x4_f32(
          false, afrag[ks], false, bfrag[ks], (short)0, acc, false, false);
    }

    // C/D layout: VGPR r -> M = r (lanes 0-15) / 8+r (lanes 16-31), N=lane%16.
    // Unconditional: one base, constant byte offsets (pad rows store 0).
    float* yp = Yout + (n << 12) + (half << 11) + col;
#pragma unroll
    for (int r = 0; r < 8; ++r) yp[r << 8] = acc[r] + bv[r];
  }
}

// BatchNorm1d training mode (+optional ReLU), in place on padded Y
// [64][16][256].  Wave w < CH reduces channel w (deterministic order);
// waves CH..15 emit {0,0} so the apply sweep is uniform and keeps pads at 0.
template <int CH>
__device__ void bn_apply(float* __restrict__ Y,
                         const float* __restrict__ g,
                         const float* __restrict__ bb, bool relu,
                         float* __restrict__ sstats) {
  const int tid  = threadIdx.x;
  const int wave = tid >> 5;
  const int lane = tid & 31;
  if (wave < CH) {
    float s = 0.0f, s2 = 0.0f;
    for (int idx = lane; idx < 16384; idx += 32) {
      const int n = idx >> 8, l = idx & 255;
      const float v = Y[(n << 12) + (wave << 8) + l];
      s += v;
      s2 += v * v;
    }
#pragma unroll
    for (int off = 16; off > 0; off >>= 1) {
      s  += __shfl_down(s, off);
      s2 += __shfl_down(s2, off);
    }
    if (lane == 0) {
      const float mean = s * (1.0f / 16384.0f);
      const float var  = s2 * (1.0f / 16384.0f) - mean * mean;  // biased var
      const float sc   = g[wave] / sqrtf(var + 1e-5f);
      sstats[wave * 2 + 0] = sc;
      sstats[wave * 2 + 1] = bb[wave] - mean * sc;
    }
  } else if (wave < 16) {
    if (lane == 0) { sstats[wave * 2] = 0.0f; sstats[wave * 2 + 1] = 0.0f; }
  }
  __syncthreads();
  for (int i = tid; i < 64 * 16 * 256; i += 1024) {
    const int ch = (i >> 8) & 15;
    float v = Y[i] * sstats[ch * 2] + sstats[ch * 2 + 1];
    if (relu) v = fmaxf(v, 0.0f);
    Y[i] = v;
  }
  gsync();
}

// --- zero-padded grouped weight matrices -> LDS A [16][KPAD] row-major ------
__device__ void buildA_se(const float* __restrict__ w, float* __restrict__ A) {
  // (12,1,3), groups=3: out ch m uses input ch m/4.  K = 3ic*3tap = 9, pad 12.
  for (int i = threadIdx.x; i < 16 * 12; i += 1024) {
    const int m = i / 12, kk = i % 12;
    float v = 0.0f;
    if (m < 12 && kk < 9) {
      const int ic = kk / 3, k = kk % 3;
      if (ic == (m >> 2)) v = w[m * 3 + k];
    }
    A[i] = v;
  }
}
__device__ void buildA_hdf(const float* __restrict__ w, float* __restrict__ A) {
  // (12,4,3), groups=3: out m uses in (m/4)*4 .. +3.  K = 12*3 = 36.
  for (int i = threadIdx.x; i < 16 * 36; i += 1024) {
    const int m = i / 36, kk = i % 36;
    float v = 0.0f;
    if (m < 12) {
      const int ic = kk / 3, k = kk % 3;
      if ((ic >> 2) == (m >> 2)) v = w[(m * 4 + (ic & 3)) * 3 + k];
    }
    A[i] = v;
  }
}
__device__ void buildA_comp(const float* __restrict__ w, float* __restrict__ A) {
  // (3,4,3), groups=3: out m uses in m*4 .. +3.  K = 36.
  for (int i = threadIdx.x; i < 16 * 36; i += 1024) {
    const int m = i / 36, kk = i % 36;
    float v = 0.0f;
    if (m < 3) {
      const int ic = kk / 3, k = kk % 3;
      if ((ic >> 2) == m) v = w[(m * 4 + (ic & 3)) * 3 + k];
    }
    A[i] = v;
  }
}
__device__ void buildA_pw(const float* __restrict__ w, float* __restrict__ A) {
  // (3,3,1), dense 1x1.  K = 3, pad 4.
  for (int i = threadIdx.x; i < 16 * 4; i += 1024) {
    const int m = i / 4, kk = i % 4;
    A[i] = (m < 3 && kk < 3) ? w[m * 3 + kk] : 0.0f;
  }
}

__global__ __launch_bounds__(1024) void net_kernel(
    const unsigned* __restrict__ hist, float* P0, float* P1, float* YB,
    float* ZB, NetParams np, float* __restrict__ out) {
  __shared__ float Alds[16 * 36];
  __shared__ float sstats[32];
  const int tid = threadIdx.x;

  // hist counts -> padded float input [64][16][256] (pad channels = 0).
  for (int i = tid; i < 64 * 16 * 256; i += 1024) {
    const int n = i >> 12, ch = (i >> 8) & 15, l = i & 255;
    P0[i] = (ch < 3) ? (float)hist[(n * 3 + ch) * 256 + l] : 0.0f;
  }
  gsync();

  float* cur = P0;
  float* nxt = P1;
#pragma unroll 1
  for (int s = 0; s < 4; ++s) {
    const StageParams& p = np.s[s];

    buildA_se(p.se_w, Alds);
    __syncthreads();
    conv_wmma<3, 12, 12>(cur, YB, p.se_b, Alds);
    gsync();
    bn_apply<12>(YB, p.se_g, p.se_bb, true, sstats);

    buildA_hdf(p.hdf_w, Alds);
    __syncthreads();
    conv_wmma<3, 36, 12>(YB, ZB, p.hdf_b, Alds);
    gsync();
    bn_apply<12>(ZB, p.hdf_g, p.hdf_bb, true, sstats);

    buildA_comp(p.comp_w, Alds);
    __syncthreads();
    conv_wmma<3, 36, 3>(ZB, YB, p.comp_b, Alds);   // reuse YB
    gsync();
    bn_apply<3>(YB, p.comp_g, p.comp_bb, false, sstats);   // no ReLU

    buildA_pw(p.pw_w, Alds);
    __syncthreads();
    conv_wmma<1, 4, 3>(YB, nxt, p.pw_b, Alds);
    gsync();
    bn_apply<3>(nxt, p.pw_g, p.pw_bb, true, sstats);

    float* t = cur; cur = nxt; nxt = t;
  }

  // padded [64][16][256] -> [64,3,256].reshape(64,768) -> [64,1,768,1].
  for (int i = tid; i < 49152; i += 1024) {
    const int t = i >> 8;          // b*3 + c
    const int b = t / 3, c = t - b * 3, l = i & 255;
    out[i] = cur[(b << 12) + (c << 8) + l];
  }
}

// ---------------------------------------------------------------------------
// Host side
// ---------------------------------------------------------------------------
extern "C" void kernel_launch(void* const* d_in, const int* in_sizes, int n_in,
                              void* d_out, int out_size, void* d_ws,
                              size_t ws_size, hipStream_t stream) {
  const float* img = (const float*)d_in[0];

  char* ws = (char*)d_ws;
  const size_t SZH = 64u * 3u * 256u * 4u;      // 196608 B (hist)
  const size_t SZP = 64u * 16u * 256u * 4u;     // 1 MB (padded tensor)
  unsigned* hist = (unsigned*)ws;
  float* P0 = (float*)(ws + 262144);
  float* P1 = (float*)(ws + 262144 + SZP);
  float* YB = (float*)(ws + 262144 + 2 * SZP);
  float* ZB = (float*)(ws + 262144 + 3 * SZP);

  // The harness may flatten each stage's param dict in insertion order
  // (se_w first, 36 elems) or JAX-sorted key order (comp_b first, 3 elems).
  const bool sorted_keys = (n_in > 1 && in_sizes[1] == 3);
  NetParams np;
  for (int s = 0; s < 4; ++s) {
    const float* L[16];
    for (int j = 0; j < 16; ++j) L[j] = (const float*)d_in[1 + s * 16 + j];
    StageParams& sp = np.s[s];
    if (!sorted_keys) {
      sp.se_w = L[0];  sp.se_b = L[1];  sp.se_g = L[2];  sp.se_bb = L[3];
      sp.hdf_w = L[4]; sp.hdf_b = L[5]; sp.hdf_g = L[6]; sp.hdf_bb = L[7];
      sp.comp_w = L[8]; sp.comp_b = L[9]; sp.comp_g = L[10]; sp.comp_bb = L[11];
      sp.pw_w = L[12]; sp.pw_b = L[13]; sp.pw_g = L[14]; sp.pw_bb = L[15];
    } else {
      sp.comp_b = L[0]; sp.comp_bb = L[1]; sp.comp_g = L[2]; sp.comp_w = L[3];
      sp.hdf_b = L[4];  sp.hdf_bb = L[5]; sp.hdf_g = L[6];  sp.hdf_w = L[7];
      sp.pw_b = L[8];   sp.pw_bb = L[9]; sp.pw_g = L[10];   sp.pw_w = L[11];
      sp.se_b = L[12];  sp.se_bb = L[13]; sp.se_g = L[14];  sp.se_w = L[15];
    }
  }

  // Accumulator histogram must start at zero every call (harness poisons ws
  // once and never re-poisons; memset node is graph-capturable).
  hipMemsetAsync(hist, 0, SZH, stream);

  hist_kernel<<<dim3(192 * HSPLIT), dim3(256), 0, stream>>>(img, hist);
  net_kernel<<<dim3(1), dim3(1024), 0, stream>>>(hist, P0, P1, YB, ZB, np,
                                                 (float*)d_out);
}